// SRHGNLayerPlus_33294586479049
// MI455X (gfx1250) — compile-verified
//
#include <hip/hip_runtime.h>
#include <math.h>

#define N_A 200000
#define N_P 400000
#define N_S 5000
#define DIM 128
#define NEDGE 1000000

typedef float v2f __attribute__((ext_vector_type(2)));
typedef float v8f __attribute__((ext_vector_type(8)));

// ---------------- helpers ----------------
__device__ __forceinline__ float wave_sum32(float v) {
#pragma unroll
  for (int off = 16; off > 0; off >>= 1) v += __shfl_xor(v, off, 32);
  return v;
}
__device__ __forceinline__ float lrelu(float x) { return x > 0.f ? x : 0.01f * x; }
__device__ __forceinline__ float gelu_exact(float x) {
  return 0.5f * x * (1.f + erff(x * 0.70710678118654752440f));
}
// monotone order-preserving float<->uint encoding for atomicMax on floats
__device__ __forceinline__ unsigned f2ord(float f) {
  unsigned b = __float_as_uint(f);
  return (b & 0x80000000u) ? ~b : (b | 0x80000000u);
}
__device__ __forceinline__ float ord2f(unsigned u) {
  return (u & 0x80000000u) ? __uint_as_float(u & 0x7FFFFFFFu) : __uint_as_float(~u);
}
__device__ __forceinline__ void ld4(const float* p, float* d) {
  float4 v = *(const float4*)p; d[0] = v.x; d[1] = v.y; d[2] = v.z; d[3] = v.w;
}
__device__ __forceinline__ void st4(float* p, const float* d) {
  float4 v; v.x = d[0]; v.y = d[1]; v.z = d[2]; v.w = d[3]; *(float4*)p = v;
}

// ---------------- WMMA GEMM: C[N x 128] = A[N x 128] * W[128 x 128] + b ----------------
// 256 threads = 8 waves; each wave computes a 16-row x 128-col strip.
// fp32 WMMA 16x16x4: exact fp32 accumulate (matches reference numerics).
__global__ __launch_bounds__(256) void gemm_nd128(
    const float* __restrict__ A, const float* __restrict__ W,
    const float* __restrict__ bias, float* __restrict__ C, int N) {
  __shared__ float sW[DIM * DIM];  // 64 KB: whole weight staged once per block
  const int t = threadIdx.x;
  {
    const float4* Wv = (const float4*)W;
    float4* sWv = (float4*)sW;
#pragma unroll
    for (int i = 0; i < 16; ++i) sWv[t + i * 256] = Wv[t + i * 256];
  }
  __syncthreads();

  const int wave = t >> 5, lane = t & 31;
  const long row0 = (long)blockIdx.x * 128 + wave * 16;
  if (row0 >= N) return;
  const int m  = lane & 15;   // row within 16-tile / col within 16-tile
  const int hi = lane >> 4;   // 0 = low half-wave, 1 = high half-wave
  long rowA = row0 + m;
  if (rowA >= N) rowA = N - 1;  // clamp: EXEC must be all-1s for WMMA; stores guarded
  const float* Arow = A + rowA * (long)DIM;

  v8f acc[8] = {};  // 8 N-tiles of 16x16 f32 accumulators

#pragma unroll 4
  for (int k0 = 0; k0 < DIM; k0 += 4) {
    // A frag (16x4 f32): v0 = {K=k0 lo half, K=k0+2 hi half}, v1 = {k0+1, k0+3}
    v2f a;
    a.x = Arow[k0 + hi * 2];
    a.y = Arow[k0 + hi * 2 + 1];
#pragma unroll
    for (int n = 0; n < 8; ++n) {
      const int col = n * 16 + m;
      // B frag (4x16 f32): v0 = rows {k0 lo, k0+1 hi}, v1 = rows {k0+2 lo, k0+3 hi}
      v2f b;
      b.x = sW[(k0 + hi) * DIM + col];
      b.y = sW[(k0 + hi + 2) * DIM + col];
      acc[n] = __builtin_amdgcn_wmma_f32_16x16x4_f32(
          false, a, false, b, (short)0, acc[n], false, false);
    }
  }

  // D layout: VGPR r -> M = r (lanes 0-15) / M = 8+r (lanes 16-31), N = lane%16
  const int mBase = hi * 8;
#pragma unroll
  for (int n = 0; n < 8; ++n) {
    const int col = n * 16 + m;
    const float bcol = bias[col];
#pragma unroll
    for (int r = 0; r < 8; ++r) {
      const long orow = row0 + mBase + r;
      if (orow < N) C[orow * (long)DIM + col] = acc[n][r] + bcol;
    }
  }
}

// ---------------- per-node attention projection: out[n][h] = X[n,:] . W[:,h] + b[h] ----------------
__global__ void attproj_kernel(const float* __restrict__ X, const float* __restrict__ W,
                               const float* __restrict__ b, float* __restrict__ out, int N) {
  int id = blockIdx.x * blockDim.x + threadIdx.x;
  if (id >= N * 4) return;
  int n = id >> 2, h = id & 3;
  const float* row = X + (long)n * DIM;
  float acc = b[h];
#pragma unroll 8
  for (int k = 0; k < DIM; ++k) acc += row[k] * W[k * 4 + h];
  out[id] = acc;
}

// ---------------- edge softmax pass 1: per-(dst,head) running max ----------------
__global__ void edge_max_kernel(const int* __restrict__ src, const int* __restrict__ dst,
                                const float* __restrict__ as, const float* __restrict__ ad,
                                unsigned* __restrict__ amax, int E) {
  int id = blockIdx.x * blockDim.x + threadIdx.x;
  if (id >= E * 4) return;
  int e = id >> 2, h = id & 3;
  int s = src[e], d = dst[e];
  float a = lrelu(as[s * 4 + h] + ad[d * 4 + h]);
  atomicMax(&amax[d * 4 + h], f2ord(a));
}

// ---------------- edge softmax pass 2: per-(dst,head) sum of exp ----------------
__global__ void edge_sum_kernel(const int* __restrict__ src, const int* __restrict__ dst,
                                const float* __restrict__ as, const float* __restrict__ ad,
                                const unsigned* __restrict__ amax, float* __restrict__ asum, int E) {
  int id = blockIdx.x * blockDim.x + threadIdx.x;
  if (id >= E * 4) return;
  int e = id >> 2, h = id & 3;
  int s = src[e], d = dst[e];
  float a = lrelu(as[s * 4 + h] + ad[d * 4 + h]);
  atomicAdd(&asum[d * 4 + h], expf(a - ord2f(amax[d * 4 + h])));
}

// ---------------- edge pass 3: z[dst] += hs[src] * att  (wave per edge, lane = feature) ----------------
__global__ __launch_bounds__(256) void edge_msg_kernel(
    const int* __restrict__ src, const int* __restrict__ dst,
    const float* __restrict__ as, const float* __restrict__ ad,
    const unsigned* __restrict__ amax, const float* __restrict__ asum,
    const float* __restrict__ hs, float* __restrict__ z, int E) {
  int e = blockIdx.x * 8 + (threadIdx.x >> 5);
  if (e >= E) return;
  int lane = threadIdx.x & 31;
  int s = src[e], d = dst[e];
  const float* hrow = hs + (long)s * DIM;
  float* zrow = z + (long)d * DIM;
#pragma unroll
  for (int h = 0; h < 4; ++h) {
    float a = lrelu(as[s * 4 + h] + ad[d * 4 + h]);
    float att = expf(a - ord2f(amax[d * 4 + h])) / asum[d * 4 + h];
    int f = h * 32 + lane;
    atomicAdd(&zrow[f], hrow[f] * att);
  }
}

// ---------------- relation-attention logits: rel[ai][t] = rel_emb[ai] . relA_W[ai][:,t] + b ----------------
__global__ void rel_logits_kernel(const float* __restrict__ rel_emb, const float* __restrict__ relA_W,
                                  const float* __restrict__ relA_b, float* __restrict__ out) {
  int id = threadIdx.x;
  if (id >= 12) return;
  int ai = id >> 2, t = id & 3;
  float acc = relA_b[ai * 4 + t];
  for (int k = 0; k < DIM; ++k) acc += rel_emb[ai * DIM + k] * relA_W[(long)ai * DIM * 4 + k * 4 + t];
  out[id] = acc;
}

// ---------------- fusion: semantic+relation attention, gelu, L2-norm (wave per node) ----------------
__global__ __launch_bounds__(256) void fuse_kernel(
    const float* __restrict__ z0, const float* __restrict__ z1,
    const float* __restrict__ hd, const float* __restrict__ h,
    const float* __restrict__ semS_W, const float* __restrict__ semS_b,
    const float* __restrict__ semD_W, const float* __restrict__ semD_b,
    const float* __restrict__ rlog, float* __restrict__ out,
    int N, int ai0, int ai1, int R) {
  const int node = blockIdx.x * 8 + (threadIdx.x >> 5);
  if (node >= N) return;
  const int lane = threadIdx.x & 31;
  const long base = (long)node * DIM + lane * 4;

  float h4[4], hd4[4], za[4], zb[4];
  ld4(h + base, h4);
  ld4(hd + base, hd4);
  ld4(z0 + base, za);
  if (R == 2) ld4(z1 + base, zb);
  else { zb[0] = zb[1] = zb[2] = zb[3] = 0.f; }

  // L2 normalizations (for semantic logits only)
  float ss = hd4[0]*hd4[0] + hd4[1]*hd4[1] + hd4[2]*hd4[2] + hd4[3]*hd4[3];
  float inv = 1.f / fmaxf(sqrtf(wave_sum32(ss)), 1e-9f);
  float nd[4];  for (int c = 0; c < 4; ++c) nd[c] = hd4[c] * inv;
  ss = za[0]*za[0] + za[1]*za[1] + za[2]*za[2] + za[3]*za[3];
  inv = 1.f / fmaxf(sqrtf(wave_sum32(ss)), 1e-9f);
  float zna[4]; for (int c = 0; c < 4; ++c) zna[c] = za[c] * inv;
  float znb[4] = {0.f, 0.f, 0.f, 0.f};
  if (R == 2) {
    ss = zb[0]*zb[0] + zb[1]*zb[1] + zb[2]*zb[2] + zb[3]*zb[3];
    inv = 1.f / fmaxf(sqrtf(wave_sum32(ss)), 1e-9f);
    for (int c = 0; c < 4; ++c) znb[c] = zb[c] * inv;
  }

  // semantic logits: s_r[t] = zn_r . semS_W[ai_r][:,t] + nd . semD_W[ai_r][:,t] + biases
  float s0[4], s1[4] = {0.f, 0.f, 0.f, 0.f};
  {
    const float* WS = semS_W + (long)ai0 * DIM * 4 + (long)(lane * 4) * 4;
    const float* WD = semD_W + (long)ai0 * DIM * 4 + (long)(lane * 4) * 4;
    float p[4] = {0.f, 0.f, 0.f, 0.f};
#pragma unroll
    for (int c = 0; c < 4; ++c)
#pragma unroll
      for (int t2 = 0; t2 < 4; ++t2) p[t2] += zna[c] * WS[c * 4 + t2] + nd[c] * WD[c * 4 + t2];
#pragma unroll
    for (int t2 = 0; t2 < 4; ++t2)
      s0[t2] = wave_sum32(p[t2]) + semS_b[ai0 * 4 + t2] + semD_b[ai0 * 4 + t2];
  }
  if (R == 2) {
    const float* WS = semS_W + (long)ai1 * DIM * 4 + (long)(lane * 4) * 4;
    const float* WD = semD_W + (long)ai1 * DIM * 4 + (long)(lane * 4) * 4;
    float p[4] = {0.f, 0.f, 0.f, 0.f};
#pragma unroll
    for (int c = 0; c < 4; ++c)
#pragma unroll
      for (int t2 = 0; t2 < 4; ++t2) p[t2] += znb[c] * WS[c * 4 + t2] + nd[c] * WD[c * 4 + t2];
#pragma unroll
    for (int t2 = 0; t2 < 4; ++t2)
      s1[t2] = wave_sum32(p[t2]) + semS_b[ai1 * 4 + t2] + semD_b[ai1 * 4 + t2];
  }

  // attn = 0.5 * softmax_r(lrelu(sem)) + 0.5 * softmax_r(lrelu(rel))
  float att0[4], att1[4];
  if (R == 1) {
#pragma unroll
    for (int t2 = 0; t2 < 4; ++t2) { att0[t2] = 1.f; att1[t2] = 0.f; }
  } else {
#pragma unroll
    for (int t2 = 0; t2 < 4; ++t2) {
      float l0 = lrelu(s0[t2]), l1 = lrelu(s1[t2]);
      float mm = fmaxf(l0, l1);
      float e0 = expf(l0 - mm), e1 = expf(l1 - mm);
      float sa0 = e0 / (e0 + e1);
      float r0 = lrelu(rlog[ai0 * 4 + t2]), r1 = lrelu(rlog[ai1 * 4 + t2]);
      float rm = fmaxf(r0, r1);
      float g0 = expf(r0 - rm), g1 = expf(r1 - rm);
      float ra0 = g0 / (g0 + g1);
      att0[t2] = 0.5f * sa0 + 0.5f * ra0;
      att1[t2] = 0.5f * (1.f - sa0) + 0.5f * (1.f - ra0);
    }
  }

  const int th = lane >> 3;  // type-head index for this lane's 4 features (D/HT = 32)
  float g[4]; float gg = 0.f;
#pragma unroll
  for (int c = 0; c < 4; ++c) {
    float v = za[c] * att0[th] + zb[c] * att1[th] + h4[c];
    g[c] = gelu_exact(v);
    gg += g[c] * g[c];
  }
  float invo = 1.f / fmaxf(sqrtf(wave_sum32(gg)), 1e-9f);
#pragma unroll
  for (int c = 0; c < 4; ++c) g[c] *= invo;
  st4(out + base, g);
}

// ---------------- subject fallback: out = norm(gelu(hd_s + h_s)), in place on staged hd_s ----------------
__global__ __launch_bounds__(256) void subject_kernel(float* __restrict__ io,
                                                      const float* __restrict__ h, int N) {
  int node = blockIdx.x * 8 + (threadIdx.x >> 5);
  if (node >= N) return;
  int lane = threadIdx.x & 31;
  long base = (long)node * DIM + lane * 4;
  float a[4], b[4], g[4];
  ld4(io + base, a);
  ld4(h + base, b);
  float ss = 0.f;
#pragma unroll
  for (int c = 0; c < 4; ++c) { g[c] = gelu_exact(a[c] + b[c]); ss += g[c] * g[c]; }
  float inv = 1.f / fmaxf(sqrtf(wave_sum32(ss)), 1e-9f);
#pragma unroll
  for (int c = 0; c < 4; ++c) g[c] *= inv;
  st4(io + base, g);
}

// ---------------- driver ----------------
extern "C" void kernel_launch(void* const* d_in, const int* in_sizes, int n_in,
                              void* d_out, int out_size, void* d_ws, size_t ws_size,
                              hipStream_t stream) {
  (void)in_sizes; (void)n_in; (void)ws_size;
  const float* h_author  = (const float*)d_in[0];
  const float* h_paper   = (const float*)d_in[1];
  const float* h_subject = (const float*)d_in[2];
  const int* e_src[3] = {(const int*)d_in[3], (const int*)d_in[5], (const int*)d_in[7]};
  const int* e_dst[3] = {(const int*)d_in[4], (const int*)d_in[6], (const int*)d_in[8]};
  const float* node_W = (const float*)d_in[9];
  const float* node_b = (const float*)d_in[10];
  const float* edge_W = (const float*)d_in[11];
  const float* edge_b = (const float*)d_in[12];
  const float* srcA_W = (const float*)d_in[13];
  const float* srcA_b = (const float*)d_in[14];
  const float* dstA_W = (const float*)d_in[15];
  const float* dstA_b = (const float*)d_in[16];
  const float* semS_W = (const float*)d_in[17];
  const float* semS_b = (const float*)d_in[18];
  const float* semD_W = (const float*)d_in[19];
  const float* semD_b = (const float*)d_in[20];
  const float* relA_W = (const float*)d_in[21];
  const float* relA_b = (const float*)d_in[22];
  const float* rel_emb = (const float*)d_in[23];

  float* out = (float*)d_out;
  float* z_a = out;                               // z_r2 accumulator + final author out
  float* z_p = out + (size_t)N_A * DIM;           // z_r0 accumulator + final paper out
  float* z_s = out + (size_t)(N_A + N_P) * DIM;   // hd_s staging + final subject out

  float* p = (float*)d_ws;
  float* hd_a = p;  p += (size_t)N_A * DIM;
  float* hd_p = p;  p += (size_t)N_P * DIM;
  float* hs0  = p;  p += (size_t)N_A * DIM;
  float* hs1  = p;  p += (size_t)N_S * DIM;
  float* hs2  = p;  p += (size_t)N_P * DIM;
  float* z_r1 = p;  p += (size_t)N_P * DIM;
  float* a_src = p; p += (size_t)N_P * 4;   // reused per relation (max dst/src count)
  float* a_dst = p; p += (size_t)N_P * 4;
  unsigned* amax = (unsigned*)p; p += (size_t)N_P * 4;
  float* asum = p;  p += (size_t)N_P * 4;
  float* rlog = p;  p += 16;

  hipMemsetAsync(d_out, 0, (size_t)out_size * sizeof(float), stream);
  hipMemsetAsync(z_r1, 0, (size_t)N_P * DIM * sizeof(float), stream);

  // six dense transforms via fp32 WMMA
  const size_t WS = (size_t)DIM * DIM;
  gemm_nd128<<<dim3((N_A + 127) / 128), 256, 0, stream>>>(h_author,  node_W + 0 * WS, node_b + 0 * DIM, hd_a, N_A);
  gemm_nd128<<<dim3((N_P + 127) / 128), 256, 0, stream>>>(h_paper,   node_W + 1 * WS, node_b + 1 * DIM, hd_p, N_P);
  gemm_nd128<<<dim3((N_S + 127) / 128), 256, 0, stream>>>(h_subject, node_W + 2 * WS, node_b + 2 * DIM, z_s,  N_S);
  gemm_nd128<<<dim3((N_A + 127) / 128), 256, 0, stream>>>(h_author,  edge_W + 0 * WS, edge_b + 0 * DIM, hs0,  N_A);
  gemm_nd128<<<dim3((N_S + 127) / 128), 256, 0, stream>>>(h_subject, edge_W + 1 * WS, edge_b + 1 * DIM, hs1,  N_S);
  gemm_nd128<<<dim3((N_P + 127) / 128), 256, 0, stream>>>(h_paper,   edge_W + 2 * WS, edge_b + 2 * DIM, hs2,  N_P);

  const float* hs_r[3] = {hs0, hs1, hs2};
  const float* hd_r[3] = {hd_p, hd_p, hd_a};
  const int nsrc[3] = {N_A, N_S, N_P};
  const int ndst[3] = {N_P, N_P, N_A};
  float* zr[3] = {z_p, z_r1, z_a};

  for (int r = 0; r < 3; ++r) {
    attproj_kernel<<<dim3((nsrc[r] * 4 + 255) / 256), 256, 0, stream>>>(
        hs_r[r], srcA_W + (size_t)r * DIM * 4, srcA_b + r * 4, a_src, nsrc[r]);
    attproj_kernel<<<dim3((ndst[r] * 4 + 255) / 256), 256, 0, stream>>>(
        hd_r[r], dstA_W + (size_t)r * DIM * 4, dstA_b + r * 4, a_dst, ndst[r]);
    hipMemsetAsync(amax, 0, (size_t)ndst[r] * 4 * sizeof(unsigned), stream);
    hipMemsetAsync(asum, 0, (size_t)ndst[r] * 4 * sizeof(float), stream);
    edge_max_kernel<<<dim3((NEDGE * 4 + 255) / 256), 256, 0, stream>>>(
        e_src[r], e_dst[r], a_src, a_dst, amax, NEDGE);
    edge_sum_kernel<<<dim3((NEDGE * 4 + 255) / 256), 256, 0, stream>>>(
        e_src[r], e_dst[r], a_src, a_dst, amax, asum, NEDGE);
    edge_msg_kernel<<<dim3((NEDGE + 7) / 8), 256, 0, stream>>>(
        e_src[r], e_dst[r], a_src, a_dst, amax, asum, hs_r[r], zr[r], NEDGE);
  }

  rel_logits_kernel<<<dim3(1), 32, 0, stream>>>(rel_emb, relA_W, relA_b, rlog);

  fuse_kernel<<<dim3((N_A + 7) / 8), 256, 0, stream>>>(
      z_a, z_a, hd_a, h_author, semS_W, semS_b, semD_W, semD_b, rlog, z_a, N_A, 0, 0, 1);
  fuse_kernel<<<dim3((N_P + 7) / 8), 256, 0, stream>>>(
      z_p, z_r1, hd_p, h_paper, semS_W, semS_b, semD_W, semD_b, rlog, z_p, N_P, 1, 2, 2);
  subject_kernel<<<dim3((N_S + 7) / 8), 256, 0, stream>>>(z_s, h_subject, N_S);
}